// Novel_loss_2241972928636
// MI455X (gfx1250) — compile-verified
//
#include <hip/hip_runtime.h>
#include <math.h>

// MI455X (gfx1250) implementation of the YOLO-style loss.
//
// Roofline: the loss reduces to (a) <=256 sparse obj-cell terms and (b) a
// full-grid sum of pred_obj^2 over only channel 4 => 10.5 MB streamed once
// (~0.45us @ 23.3 TB/s). To keep it memory-bound, the hot loop uses raw
// hardware transcendentals (v_exp_f32 / v_rcp_f32, ~1ulp) for the double
// sigmoid, and the summation itself runs on the matrix unit via
// V_WMMA_F32_16X16X4_F32 with an all-ones B (D[m][n] += sum_k A[m][k]).
// All kernels are deterministic (fixed-order reductions, no atomics) and
// graph-capture safe.

typedef __attribute__((ext_vector_type(2))) float v2f;
typedef __attribute__((ext_vector_type(8))) float v8f;

#define RESQ 8.0f
#define B_ 32
#define A_ 5
#define H_ 128
#define W_ 128
#define NLAB 256
#define GRID_CELLS 2621440.0f   // B*A*H*W
#define LOG2E 1.44269504088896340736f

__device__ __forceinline__ float sigf(float x)  { return 1.0f / (1.0f + expf(-x)); }
__device__ __forceinline__ float sig2f(float x) { return sigf(sigf(x)); }

// Fast double-sigmoid: v_exp_f32 computes 2^x, so sigmoid(x) =
// v_rcp(1 + 2^(-x*log2e)). ~1ulp per hw op -> ~1e-6 relative error.
__device__ __forceinline__ float fast_sig(float x) {
    return __builtin_amdgcn_rcpf(1.0f + __builtin_amdgcn_exp2f(-x * LOG2E));
}
__device__ __forceinline__ float fast_sig2(float x) { return fast_sig(fast_sig(x)); }

// ---------------------------------------------------------------------------
// Kernel 1: build targets (argmax anchor IoU + dedupe) and obj-cell losses.
// One thread per label; last-duplicate-wins matches XLA scatter order.
// Cold path (256 threads): keep the precise libm sigmoid/exp here.
// ---------------------------------------------------------------------------
__global__ __launch_bounds__(256) void k_obj(const float* __restrict__ out,
                                             const float* __restrict__ labels,
                                             const float* __restrict__ anchors,
                                             float* __restrict__ ws)
{
    __shared__ int   keys[NLAB];
    __shared__ float red[NLAB];
    const int t = threadIdx.x;
    const float* L = labels + t * 9;
    const float bx = L[0], by = L[1], bw = L[2], bh = L[3];
    const float tr0 = L[4], tr1 = L[5], tr2 = L[6], tr3 = L[7];
    const int img = (int)L[8];

    // _build_targets anchor IoU (clip at 0.0), jnp.argmax first-wins ties.
    const float tl1x = bx - 0.5f * bw, tl1y = by - 0.5f * bh;
    const float br1x = bx + 0.5f * bw, br1y = by + 0.5f * bh;
    const float a1 = fmaxf(br1x - tl1x, 0.f) * fmaxf(br1y - tl1y, 0.f);
    int ind = 0; float best = -1e30f;
    for (int a = 0; a < A_; a++) {
        const float ax = anchors[2 * a], ay = anchors[2 * a + 1];
        // anchs = [xy - a/2, xy + a/2] then treated as (x,y,w,h) (as written)
        const float c0 = bx - 0.5f * ax, c1 = by - 0.5f * ay;
        const float c2 = bx + 0.5f * ax, c3 = by + 0.5f * ay;
        const float tl2x = c0 - 0.5f * c2, tl2y = c1 - 0.5f * c3;
        const float br2x = c0 + 0.5f * c2, br2y = c1 + 0.5f * c3;
        const float sx = fmaxf(fminf(br1x, br2x) - fmaxf(tl1x, tl2x), 0.f);
        const float sy = fmaxf(fminf(br1y, br2y) - fmaxf(tl1y, tl2y), 0.f);
        const float inter = sx * sy;
        const float a2 = fmaxf(br2x - tl2x, 0.f) * fmaxf(br2y - tl2y, 0.f);
        const float iou = inter / (a1 + a2 - inter);
        if (iou > best) { best = iou; ind = a; }
    }
    const int xc = (int)floorf(bx / RESQ);
    const int yc = (int)floorf(by / RESQ);
    const int key = ((img * A_ + ind) * H_ + yc) * W_ + xc;
    keys[t] = key;
    __syncthreads();
    bool winner = true;
    for (int j = t + 1; j < NLAB; j++) if (keys[j] == key) winner = false;

    float v_loc = 0.f, v_size = 0.f, v_vel = 0.f, v_acc = 0.f;
    float v_obj = 0.f, v_p2 = 0.f, v_cnt = 0.f;
    if (winner) {
        const int base = ((img * 45 + ind * 9) * H_ + yc) * W_ + xc;
        float o[9];
        #pragma unroll
        for (int c = 0; c < 9; c++) o[c] = out[base + c * (H_ * W_)];
        const float lpx = sig2f(o[0]) + (float)xc;
        const float lpy = sig2f(o[1]) + (float)yc;
        const float adx = anchors[2 * ind] / RESQ, ady = anchors[2 * ind + 1] / RESQ;
        const float spx = sqrtf(expf(o[2]) * adx);
        const float spy = sqrtf(expf(o[3]) * ady);
        const float pobj = sig2f(o[4]);
        const float ltx = bx / RESQ, lty = by / RESQ, ltw = bw / RESQ, lth = bh / RESQ;
        const float stx = sqrtf(ltw), sty = sqrtf(lth);
        // _elem_iou(pred*res [pixels], locs [cells]) -- exactly as reference.
        const float p1x = lpx * RESQ, p1y = lpy * RESQ, p1w = spx * RESQ, p1h = spy * RESQ;
        const float t1x = p1x - 0.5f * p1w, t1y = p1y - 0.5f * p1h;
        const float b1x = p1x + 0.5f * p1w, b1y = p1y + 0.5f * p1h;
        const float t2x = ltx - 0.5f * ltw, t2y = lty - 0.5f * lth;
        const float b2x = ltx + 0.5f * ltw, b2y = lty + 0.5f * lth;
        const float sx = fmaxf(fminf(b1x, b2x) - fmaxf(t1x, t2x), 0.f);
        const float sy = fmaxf(fminf(b1y, b2y) - fmaxf(t1y, t2y), 0.f);
        const float inter = sx * sy;
        const float s1x = fmaxf(b1x - t1x, 1e-6f), s1y = fmaxf(b1y - t1y, 1e-6f);
        const float s2x = fmaxf(b2x - t2x, 1e-6f), s2y = fmaxf(b2y - t2y, 1e-6f);
        const float iou = inter / (s1x * s1y + s2x * s2y - inter);
        v_loc  = (ltx - lpx) * (ltx - lpx) + (lty - lpy) * (lty - lpy);
        v_size = (stx - spx) * (stx - spx) + (sty - spy) * (sty - spy);
        v_vel  = (tr0 - o[5]) * (tr0 - o[5]) + (tr1 - o[6]) * (tr1 - o[6]);
        v_acc  = (tr2 - o[7]) * (tr2 - o[7]) + (tr3 - o[8]) * (tr3 - o[8]);
        v_obj  = (iou - pobj) * (iou - pobj);
        v_p2   = pobj * pobj;
        v_cnt  = 1.0f;
    }
    float vals[7] = {v_loc, v_size, v_vel, v_acc, v_obj, v_p2, v_cnt};
    for (int q = 0; q < 7; q++) {
        __syncthreads();
        red[t] = vals[q];
        __syncthreads();
        for (int s = 128; s > 0; s >>= 1) {
            if (t < s) red[t] += red[t + s];
            __syncthreads();
        }
        if (t == 0) ws[q] = red[0];
    }
}

// ---------------------------------------------------------------------------
// Kernel 2: full-grid sum of sigmoid(sigmoid(o[...,4]))^2 over channel-4
// planes only (10.5 MB). Each block owns half of one contiguous plane, so
// the plane decomposition (and div-by-5) is scalar, computed once; the 8
// loads per thread are base + k*256 (fully coalesced float4, affine).
// WMMA f32 16x16x4 with B = ones accumulates the sum in the matrix unit:
// D[m][n] += sum_k A[m][k]; column 0 holds every input value exactly once.
// No divergence -> EXEC all-ones at every WMMA.
// ---------------------------------------------------------------------------
__global__ __launch_bounds__(256) void k_noobj(const float* __restrict__ out,
                                               float* __restrict__ partials)
{
    const float4* o4p = (const float4*)out;
    const int p = blockIdx.x >> 1;            // plane 0..159 (scalar)
    const int b = p / 5;
    const int a = p - 5 * b;
    // channel-4 plane base in float4 units + half-plane select + lane
    const int base4 = (b * 45 + a * 9 + 4) * 4096
                    + (blockIdx.x & 1) * 2048 + threadIdx.x;

    v8f acc = {0.f, 0.f, 0.f, 0.f, 0.f, 0.f, 0.f, 0.f};
    const v2f bones = {1.0f, 1.0f};           // B matrix: 4x16 all-ones
    #pragma unroll
    for (int k = 0; k < 8; k++) {
        const float4 v = o4p[base4 + k * 256];
        float s0 = fast_sig2(v.x); s0 *= s0;
        float s1 = fast_sig2(v.y); s1 *= s1;
        float s2 = fast_sig2(v.z); s2 *= s2;
        float s3 = fast_sig2(v.w); s3 *= s3;
        const v2f A0 = {s0, s1};
        const v2f A1 = {s2, s3};
        acc = __builtin_amdgcn_wmma_f32_16x16x4_f32(false, A0, false, bones,
                                                    (short)0, acc, false, false);
        acc = __builtin_amdgcn_wmma_f32_16x16x4_f32(false, A1, false, bones,
                                                    (short)0, acc, false, false);
    }
    // C/D layout: VGPR r, lanes 0-15 -> M=r,N=lane ; lanes 16-31 -> M=8+r.
    // Column N=0 = lane0 regs (M=0..7) + lane16 regs (M=8..15).
    float S = acc[0] + acc[1] + acc[2] + acc[3] + acc[4] + acc[5] + acc[6] + acc[7];
    const float wave = __shfl(S, 0, 32) + __shfl(S, 16, 32);
    __shared__ float wsum[8];
    const int lane = threadIdx.x & 31;
    const int w    = threadIdx.x >> 5;
    if (lane == 0) wsum[w] = wave;
    __syncthreads();
    if (threadIdx.x == 0) {
        float tot = 0.f;
        #pragma unroll
        for (int i = 0; i < 8; i++) tot += wsum[i];
        partials[blockIdx.x] = tot;
    }
}

// ---------------------------------------------------------------------------
// Kernel 3: deterministic final combine (single wave).
// ---------------------------------------------------------------------------
__global__ __launch_bounds__(32) void k_final(const float* __restrict__ ws,
                                              float* __restrict__ outp)
{
    const int t = threadIdx.x;
    float S = 0.f;
    #pragma unroll
    for (int k = 0; k < 10; k++) S += ws[16 + t + 32 * k];   // 320 partials
    for (int off = 16; off > 0; off >>= 1) S += __shfl_down(S, off, 32);
    if (t == 0) {
        const float n_obj = ws[6];
        const float inv2n = 1.0f / (2.0f * n_obj);
        const float loc_l  = ws[0] * inv2n;
        const float size_l = ws[1] * inv2n;
        const float vel_l  = ws[2] * inv2n;
        const float acc_l  = ws[3] * inv2n;
        const float obj_l  = ws[4] / n_obj;
        // noobj: iou==0 off-object, so term = pred_obj^2 summed off-object.
        const float noobj_l = (S - ws[5]) / (GRID_CELLS - n_obj);
        outp[0] = 5.0f * (loc_l + size_l) + vel_l + acc_l + obj_l + 0.5f * noobj_l;
    }
}

extern "C" void kernel_launch(void* const* d_in, const int* in_sizes, int n_in,
                              void* d_out, int out_size, void* d_ws, size_t ws_size,
                              hipStream_t stream)
{
    const float* out     = (const float*)d_in[0];   // (32,45,128,128) f32
    const float* labels  = (const float*)d_in[1];   // (256,9) f32
    const float* anchors = (const float*)d_in[2];   // (5,2) f32
    float* ws = (float*)d_ws;                       // [0..6] obj partials, [16..336) block partials

    k_obj  <<<1,   256, 0, stream>>>(out, labels, anchors, ws);
    k_noobj<<<320, 256, 0, stream>>>(out, ws + 16);
    k_final<<<1,   32,  0, stream>>>(ws, (float*)d_out);
}